// DeeperGCN_73641509257227
// MI455X (gfx1250) — compile-verified
//
#include <hip/hip_runtime.h>
#include <math.h>

#define N_NODES 50000
#define N_EDGES 800000
#define F_INN   32
#define H       64
#define H2      128
#define E_DIMM  16
#define C_OUTN  40
#define L_LAYERS 2
#define G_GRAPHS 64
#define BN_EPS   1e-5f
#define GEN_EPS  1e-7f

typedef __attribute__((ext_vector_type(2))) float v2f;
typedef __attribute__((ext_vector_type(8))) float v8f;

#if defined(__HIP_DEVICE_COMPILE__)
#if __has_builtin(__builtin_amdgcn_wmma_f32_16x16x4_f32)
#define USE_WMMA_F32 1
#else
#define USE_WMMA_F32 0
#endif
#else
#define USE_WMMA_F32 0
#endif

// ---------- order-preserving float<->uint for atomicMax on floats ----------
__device__ __forceinline__ unsigned fkey(float f) {
  unsigned u = __float_as_uint(f);
  return (u & 0x80000000u) ? ~u : (u | 0x80000000u);
}
__device__ __forceinline__ float funkey(unsigned u) {
  return (u & 0x80000000u) ? __uint_as_float(u & 0x7FFFFFFFu) : __uint_as_float(~u);
}
#define NEG_INF_KEY 0x007FFFFFu  // fkey(-inf)

// ---------------- WMMA f32 GEMM: C[M,NT*16] = A[M,K] @ W[K,NT*16] + bias ----
// One wave computes one 16-row strip across all NT column tiles.
template <int K, int NT>
__global__ __launch_bounds__(256) void gemm_bias_kernel(
    const float* __restrict__ A, const float* __restrict__ W,
    const float* __restrict__ bias, float* __restrict__ C, int M) {
  const int NC = NT * 16;
  const int wave = threadIdx.x >> 5;
  const int lane = threadIdx.x & 31;
  const int strip = blockIdx.x * 8 + wave;
  const int nStrips = M >> 4;
  if (strip >= nStrips) return;          // wave-uniform: EXEC stays all-1s
  const int half = lane >> 4;            // 0/1 lane half
  const int lr = lane & 15;
#if USE_WMMA_F32
  v8f acc[NT] = {};
  const int row = strip * 16 + lr;       // A fragment row for this lane
  for (int k0 = 0; k0 < K; k0 += 4) {
    v2f a;
    a.x = A[row * K + k0 + 2 * half + 0];
    a.y = A[row * K + k0 + 2 * half + 1];
#pragma unroll
    for (int nt = 0; nt < NT; ++nt) {
      const int col = nt * 16 + lr;
      v2f b;
      b.x = W[(k0 + 2 * half + 0) * NC + col];
      b.y = W[(k0 + 2 * half + 1) * NC + col];
      acc[nt] = __builtin_amdgcn_wmma_f32_16x16x4_f32(
          false, a, false, b, (short)0, acc[nt], false, false);
    }
  }
#pragma unroll
  for (int nt = 0; nt < NT; ++nt) {
    const int col = nt * 16 + lr;
    const float bv = bias[col];
#pragma unroll
    for (int v = 0; v < 8; ++v)
      C[(strip * 16 + half * 8 + v) * NC + col] = acc[nt][v] + bv;
  }
#else
  // scalar fallback with identical output mapping
  for (int nt = 0; nt < NT; ++nt) {
    const int col = nt * 16 + lr;
    for (int v = 0; v < 8; ++v) {
      const int row = strip * 16 + half * 8 + v;
      float s = bias[col];
      for (int k = 0; k < K; ++k) s += A[row * K + k] * W[k * NC + col];
      C[row * NC + col] = s;
    }
  }
#endif
}

// ---------------- BatchNorm statistics (population mean/var) ---------------
__global__ __launch_bounds__(256) void bn_stats_kernel(
    const float* __restrict__ x, float* __restrict__ sums, int total, int HH) {
  __shared__ float ls[256], ls2[256];
  const int tid = threadIdx.x;
  const int stride = gridDim.x * 256;    // multiple of HH (HH in {64,128})
  float s = 0.f, s2 = 0.f;
  for (int i = blockIdx.x * 256 + tid; i < total; i += stride) {
    const float v = x[i];
    s += v; s2 += v * v;
  }
  ls[tid] = s; ls2[tid] = s2;
  __syncthreads();
  if (tid < HH) {
    for (int j = tid + HH; j < 256; j += HH) { s += ls[j]; s2 += ls2[j]; }
    atomicAdd(&sums[tid], s);
    atomicAdd(&sums[HH + tid], s2);
  }
}

__global__ void bn_finalize_kernel(const float* __restrict__ sums,
                                   float* __restrict__ mis, float inv_n, int HH) {
  const int c = threadIdx.x;
  if (c >= HH) return;
  const float mean = sums[c] * inv_n;
  const float ex2 = sums[HH + c] * inv_n;
  const float var = ex2 - mean * mean;
  mis[c] = mean;
  mis[HH + c] = 1.0f / sqrtf(var + BN_EPS);
}

__global__ void bn_relu_kernel(const float* __restrict__ x,
                               const float* __restrict__ gamma,
                               const float* __restrict__ beta,
                               const float* __restrict__ mis,
                               float* __restrict__ out, int total, int HH) {
  const int i = blockIdx.x * 256 + threadIdx.x;
  if (i >= total) return;
  const int c = i % HH;
  const float v = (x[i] - mis[c]) * mis[HH + c] * gamma[c] + beta[c];
  out[i] = fmaxf(v, 0.0f);
}

// ---------------- segment-softmax scaffolding ------------------------------
__global__ void seg_init_kernel(unsigned* __restrict__ segmax,
                                float* __restrict__ denom,
                                float* __restrict__ num, int n) {
  const int i = blockIdx.x * 256 + threadIdx.x;
  if (i >= n) return;
  segmax[i] = NEG_INF_KEY;
  denom[i] = 0.0f;
  num[i] = 0.0f;
}

// Pass 1: atomicMax of logits per destination. 4 edges x 64 features / block.
__global__ __launch_bounds__(256) void edge_max_kernel(
    const int* __restrict__ ei, const float* __restrict__ ea,
    const float* __restrict__ z, const float* __restrict__ Wl,
    const float* __restrict__ bl, const float* __restrict__ tparam,
    unsigned* __restrict__ segmax) {
  __shared__ float sW[E_DIMM * H];
  __shared__ float sEA[4 * E_DIMM];
  __shared__ int sSrc[4], sDst[4];
  const int tid = threadIdx.x;
  for (int i = tid; i < E_DIMM * H; i += 256) sW[i] = Wl[i];
  const int e0 = blockIdx.x * 4;
  if (tid < 4 * E_DIMM) {
    const int e = e0 + tid / E_DIMM;
    sEA[tid] = (e < N_EDGES) ? ea[e * E_DIMM + (tid % E_DIMM)] : 0.0f;
  }
  if (tid < 4) {
    const int e = e0 + tid;
    sSrc[tid] = (e < N_EDGES) ? ei[e] : 0;
    sDst[tid] = (e < N_EDGES) ? ei[N_EDGES + e] : 0;
  }
  __syncthreads();
  const int le = tid >> 6;
  const int c = tid & (H - 1);
  if (e0 + le >= N_EDGES) return;
  float emb = bl[c];
#pragma unroll
  for (int k = 0; k < E_DIMM; ++k) emb += sEA[le * E_DIMM + k] * sW[k * H + c];
  float m = z[sSrc[le] * H + c] + emb;
  m = fmaxf(m, 0.0f) + GEN_EPS;
  const float logit = m * tparam[0];
  atomicMax(&segmax[sDst[le] * H + c], fkey(logit));
}

// Pass 2: accumulate exp and exp*m per destination.
__global__ __launch_bounds__(256) void edge_sum_kernel(
    const int* __restrict__ ei, const float* __restrict__ ea,
    const float* __restrict__ z, const float* __restrict__ Wl,
    const float* __restrict__ bl, const float* __restrict__ tparam,
    const unsigned* __restrict__ segmax, float* __restrict__ denom,
    float* __restrict__ num) {
  __shared__ float sW[E_DIMM * H];
  __shared__ float sEA[4 * E_DIMM];
  __shared__ int sSrc[4], sDst[4];
  const int tid = threadIdx.x;
  for (int i = tid; i < E_DIMM * H; i += 256) sW[i] = Wl[i];
  const int e0 = blockIdx.x * 4;
  if (tid < 4 * E_DIMM) {
    const int e = e0 + tid / E_DIMM;
    sEA[tid] = (e < N_EDGES) ? ea[e * E_DIMM + (tid % E_DIMM)] : 0.0f;
  }
  if (tid < 4) {
    const int e = e0 + tid;
    sSrc[tid] = (e < N_EDGES) ? ei[e] : 0;
    sDst[tid] = (e < N_EDGES) ? ei[N_EDGES + e] : 0;
  }
  __syncthreads();
  const int le = tid >> 6;
  const int c = tid & (H - 1);
  if (e0 + le >= N_EDGES) return;
  float emb = bl[c];
#pragma unroll
  for (int k = 0; k < E_DIMM; ++k) emb += sEA[le * E_DIMM + k] * sW[k * H + c];
  float m = z[sSrc[le] * H + c] + emb;
  m = fmaxf(m, 0.0f) + GEN_EPS;
  const float logit = m * tparam[0];
  const int idx = sDst[le] * H + c;
  const float mx = funkey(segmax[idx]);
  const float ex = expf(logit - mx);
  atomicAdd(&denom[idx], ex);
  atomicAdd(&num[idx], ex * m);
}

// ---------------- elementwise helpers --------------------------------------
__global__ void agg_combine_kernel(const float* __restrict__ num,
                                   const float* __restrict__ denom,
                                   float* __restrict__ z, int total) {
  const int i = blockIdx.x * 256 + threadIdx.x;
  if (i >= total) return;
  z[i] = num[i] / (denom[i] + 1e-16f) + z[i];
}

__global__ void add_kernel(float* __restrict__ h, const float* __restrict__ c,
                           int total) {
  const int i = blockIdx.x * 256 + threadIdx.x;
  if (i >= total) return;
  h[i] += c[i];
}

__global__ void zero_kernel(float* __restrict__ p, int n) {
  const int i = blockIdx.x * 256 + threadIdx.x;
  if (i < n) p[i] = 0.0f;
}

// ---------------- pooling + output head ------------------------------------
__global__ void pool_kernel(const float* __restrict__ h,
                            const int* __restrict__ batch,
                            float* __restrict__ pooled,
                            float* __restrict__ counts, int total) {
  const int i = blockIdx.x * 256 + threadIdx.x;
  if (i >= total) return;
  const int n = i >> 6;      // H = 64
  const int c = i & (H - 1);
  const int b = batch[n];
  atomicAdd(&pooled[b * H + c], h[i]);
  if (c == 0) atomicAdd(&counts[b], 1.0f);
}

__global__ __launch_bounds__(64) void head_kernel(
    const float* __restrict__ pooled, const float* __restrict__ counts,
    const float* __restrict__ Wout, const float* __restrict__ bout,
    float* __restrict__ out) {
  __shared__ float sp[H];
  const int g = blockIdx.x, t = threadIdx.x;
  const float cnt = fmaxf(counts[g], 1.0f);
  sp[t] = fmaxf(pooled[g * H + t] / cnt, 0.0f);
  __syncthreads();
  if (t < C_OUTN) {
    float acc = bout[t];
#pragma unroll
    for (int c = 0; c < H; ++c) acc += sp[c] * Wout[c * C_OUTN + t];
    out[g * C_OUTN + t] = acc;
  }
}

// ---------------- driver ----------------------------------------------------
extern "C" void kernel_launch(void* const* d_in, const int* in_sizes, int n_in,
                              void* d_out, int out_size, void* d_ws,
                              size_t ws_size, hipStream_t stream) {
  (void)in_sizes; (void)n_in; (void)out_size; (void)ws_size;
  const float* x        = (const float*)d_in[0];
  const int*   ei       = (const int*)d_in[1];
  const float* eattr    = (const float*)d_in[2];
  const int*   batch    = (const int*)d_in[3];
  const float* lin_in_w = (const float*)d_in[4];
  const float* lin_in_b = (const float*)d_in[5];
  const float* ng       = (const float*)d_in[6];
  const float* nb       = (const float*)d_in[7];
  const float* ew       = (const float*)d_in[8];
  const float* eb       = (const float*)d_in[9];
  const float* tpar     = (const float*)d_in[10];
  const float* w1       = (const float*)d_in[11];
  const float* b1       = (const float*)d_in[12];
  const float* g1       = (const float*)d_in[13];
  const float* be1      = (const float*)d_in[14];
  const float* w2       = (const float*)d_in[15];
  const float* b2       = (const float*)d_in[16];
  const float* wout     = (const float*)d_in[17];
  const float* bout     = (const float*)d_in[18];
  float* out = (float*)d_out;

  // workspace layout (floats)
  float* base = (float*)d_ws;
  size_t o = 0;
  float* h  = base + o;                 o += (size_t)N_NODES * H;
  float* z  = base + o;                 o += (size_t)N_NODES * H;
  float* h1 = base + o;                 o += (size_t)N_NODES * H2;
  unsigned* segmax = (unsigned*)(base + o); o += (size_t)N_NODES * H;
  float* denom = base + o;              o += (size_t)N_NODES * H;
  float* numb  = base + o;              o += (size_t)N_NODES * H;  // + conv out
  float* sums  = base + o;              o += 2 * H2;
  float* mis   = base + o;              o += 2 * H2;
  float* pooled = base + o;             o += (size_t)G_GRAPHS * H;
  float* counts = base + o;             o += G_GRAPHS;           // right after pooled

  const int NH  = N_NODES * H;   // 3,200,000
  const int NH2 = N_NODES * H2;  // 6,400,000
  const int ewb  = (NH + 255) / 256;
  const int ewb2 = (NH2 + 255) / 256;
  const int gemm_blocks = (N_NODES / 16 + 7) / 8;  // 391
  const float inv_n = 1.0f / (float)N_NODES;

  // h = x @ Win + b            [N,32]x[32,64]
  gemm_bias_kernel<F_INN, H / 16>
      <<<gemm_blocks, 256, 0, stream>>>(x, lin_in_w, lin_in_b, h, N_NODES);

  for (int l = 0; l < L_LAYERS; ++l) {
    // z = relu(BN(h))
    zero_kernel<<<1, 256, 0, stream>>>(sums, 2 * H);
    bn_stats_kernel<<<512, 256, 0, stream>>>(h, sums, NH, H);
    bn_finalize_kernel<<<1, 256, 0, stream>>>(sums, mis, inv_n, H);
    bn_relu_kernel<<<ewb, 256, 0, stream>>>(h, ng + l * H, nb + l * H, mis, z, NH, H);

    // segment softmax aggregation (two atomic passes, emb recomputed in LDS)
    seg_init_kernel<<<ewb, 256, 0, stream>>>(segmax, denom, numb, NH);
    edge_max_kernel<<<N_EDGES / 4, 256, 0, stream>>>(
        ei, eattr, z, ew + l * E_DIMM * H, eb + l * H, tpar + l, segmax);
    edge_sum_kernel<<<N_EDGES / 4, 256, 0, stream>>>(
        ei, eattr, z, ew + l * E_DIMM * H, eb + l * H, tpar + l, segmax, denom, numb);
    agg_combine_kernel<<<ewb, 256, 0, stream>>>(numb, denom, z, NH);  // z = agg + z

    // GENConv MLP: h1 = z@w1+b1; h1 = relu(BN(h1)); c = h1@w2+b2; h += c
    gemm_bias_kernel<H, H2 / 16>
        <<<gemm_blocks, 256, 0, stream>>>(z, w1 + l * H * H2, b1 + l * H2, h1, N_NODES);
    zero_kernel<<<1, 256, 0, stream>>>(sums, 2 * H2);
    bn_stats_kernel<<<512, 256, 0, stream>>>(h1, sums, NH2, H2);
    bn_finalize_kernel<<<1, 256, 0, stream>>>(sums, mis, inv_n, H2);
    bn_relu_kernel<<<ewb2, 256, 0, stream>>>(h1, g1 + l * H2, be1 + l * H2, mis, h1, NH2, H2);
    gemm_bias_kernel<H2, H / 16>
        <<<gemm_blocks, 256, 0, stream>>>(h1, w2 + l * H2 * H, b2 + l * H, numb, N_NODES);
    add_kernel<<<ewb, 256, 0, stream>>>(h, numb, NH);
  }

  // global mean pool + output linear
  zero_kernel<<<(G_GRAPHS * H + G_GRAPHS + 255) / 256, 256, 0, stream>>>(
      pooled, G_GRAPHS * H + G_GRAPHS);
  pool_kernel<<<ewb, 256, 0, stream>>>(h, batch, pooled, counts, NH);
  head_kernel<<<G_GRAPHS, 64, 0, stream>>>(pooled, counts, wout, bout, out);
}